// KalmanFilterExact_14998025797886
// MI455X (gfx1250) — compile-verified
//
#include <hip/hip_runtime.h>

// -------- Steady-state Kalman filter via blocked parallel scan ------------
// x_t = x_{t-1} @ A_eff^T + y_t @ K^T, emitted for all t.
// Phase 1: U = Y @ K^T  (parallel f32 WMMA GEMM, stored into d_out)
// Phase 2: per-chunk local scans (zero carry) -> chunk-end states E[g]
// Phase 3: short serial carry chain using M = A_eff^CHUNK (log-doubled)
// Phase 4: per-chunk scans seeded with true carries, overwrite d_out in place
// Matrix math: V_WMMA_F32_16X16X4_F32 (exact fp32 on the precision-critical
// sequential recurrence). A_eff staged into 260 KB of LDS (CDNA5 320KB/WGP)
// with GLOBAL_LOAD_ASYNC_TO_LDS_B128 (ASYNCcnt-tracked, no VGPR staging).

typedef float v2f __attribute__((ext_vector_type(2)));
typedef float v8f __attribute__((ext_vector_type(8)));

#define NS    256           // state dim
#define NO    128           // obs dim
#define BATCH 64
#define TLEN  1024
#define CHUNK 32
#define NCH   (TLEN / CHUNK)   // 32 chunks
#define BT    16               // batch rows per workgroup
#define NBT   (BATCH / BT)     // 4 batch tiles
#define LDP   260              // padded LDS row stride (floats): banks conflict-free

__device__ __forceinline__ v8f wmma4(v2f a, v2f b, v8f c) {
  // D = A(16x4,f32) x B(4x16,f32) + C(16x16,f32)
  return __builtin_amdgcn_wmma_f32_16x16x4_f32(false, a, false, b, (short)0, c,
                                               false, false);
}

// Async direct-to-LDS stage of a row-major [NS][NS] f32 matrix into LDS rows
// padded to LDP floats.  16B per lane per issue; 64 issues per lane covers
// 256 KB with 256 threads.  Each wave drains its own ASYNCcnt; the caller's
// __syncthreads() makes the LDS contents visible workgroup-wide.
__device__ __forceinline__ void async_stage_matrix(const float* __restrict__ g,
                                                   float* sA) {
  const int tid = threadIdx.x;  // 256 threads
  for (int it = 0; it < 64; ++it) {
    int e4   = it * 256 + tid;       // 16-byte quad index, 0..16383
    int row  = e4 >> 6;              // 64 quads per 256-float row
    int col4 = e4 & 63;
    unsigned lds = (unsigned)(size_t)(sA + row * LDP + col4 * 4);
    unsigned long long ga =
        (unsigned long long)(size_t)(g + (size_t)row * NS + col4 * 4);
    asm volatile("global_load_async_to_lds_b128 %0, %1, off"
                 :: "v"(lds), "v"(ga) : "memory");
  }
  asm volatile("s_wait_asynccnt 0x0" ::: "memory");
}

// Prefetch the next step's 16x256 additive block (rows strided by add_rs)
// into the WGP cache while the WMMA loop runs.  256 threads x 64B = 16 KB.
__device__ __forceinline__ void prefetch_add(const float* p, size_t add_rs) {
  int r = threadIdx.x >> 4;          // 16 rows
  int c = threadIdx.x & 15;          // 16 x 16-float (64B) segments per row
  __builtin_prefetch(p + (size_t)r * add_rs + c * 16, 0, 0);
}

// ---------------------------------------------------------------------------
// Small helper GEMMs for A_eff and its powers (tiny: <= 33 MFLOP each).
// out = alpha * X@Y + (add ? add : 0)
__global__ void gemm_naive(float* __restrict__ out, const float* __restrict__ X,
                           const float* __restrict__ Y, int M, int N, int Kd,
                           float alpha, const float* __restrict__ add) {
  int idx = blockIdx.x * blockDim.x + threadIdx.x;
  if (idx >= M * N) return;
  int i = idx / N, j = idx % N;
  float acc = 0.f;
  for (int k = 0; k < Kd; ++k) acc += X[i * Kd + k] * Y[k * N + j];
  out[idx] = alpha * acc + (add ? add[idx] : 0.f);
}

// ---------------------------------------------------------------------------
// U[row][n] = sum_o y[row][o] * K[n][o]   (row = b*T + t), via f32 WMMA.
__global__ __launch_bounds__(256) void proj_u(float* __restrict__ U,
                                              const float* __restrict__ y,
                                              const float* __restrict__ Kmat) {
  const int row0 = blockIdx.x * 16;       // M-tile over flattened [B*T]
  const int w  = threadIdx.x >> 5;        // wave 0..7 -> two N-tiles each
  const int l  = threadIdx.x & 31;
  const int lr = l & 15, lh = l >> 4;
  const int n0a = w * 32, n0b = n0a + 16;
  v8f acc0 = {}, acc1 = {};
  const float* yrow = y + (size_t)(row0 + lr) * NO;
  const float* krA  = Kmat + (size_t)(n0a + lr) * NO;
  const float* krB  = Kmat + (size_t)(n0b + lr) * NO;
  for (int k0 = 0; k0 < NO; k0 += 4) {
    v2f a  = *(const v2f*)(yrow + k0 + 2 * lh);
    v2f b0 = *(const v2f*)(krA  + k0 + 2 * lh);
    v2f b1 = *(const v2f*)(krB  + k0 + 2 * lh);
    acc0 = wmma4(a, b0, acc0);
    acc1 = wmma4(a, b1, acc1);
  }
#pragma unroll
  for (int r = 0; r < 8; ++r) {
    size_t row = (size_t)(row0 + r + 8 * lh);
    U[row * NS + n0a + lr] = acc0[r];
    U[row * NS + n0b + lr] = acc1[r];
  }
}

// ---------------------------------------------------------------------------
// One recurrence step: sX[16,256] <- sX @ sA^T + addend ; optional global out.
// sA row-major [NS][LDP]; B-frag B[k][n] = sA[n][k] -> contiguous float2 read.
__device__ __forceinline__ void kf_step(const float* sA, float* sX,
                                        const float* addrow0, size_t add_rs,
                                        float* outrow0, size_t out_rs) {
  const int w  = threadIdx.x >> 5;
  const int l  = threadIdx.x & 31;
  const int lr = l & 15, lh = l >> 4;
  const int n0a = w * 32, n0b = n0a + 16;
  v8f acc0 = {}, acc1 = {};
  const float* xr = sX + lr * LDP;
  const float* bA = sA + (n0a + lr) * LDP;
  const float* bB = sA + (n0b + lr) * LDP;
#pragma unroll 4
  for (int k0 = 0; k0 < NS; k0 += 4) {
    v2f a  = *(const v2f*)(xr + k0 + 2 * lh);
    v2f b0 = *(const v2f*)(bA + k0 + 2 * lh);
    v2f b1 = *(const v2f*)(bB + k0 + 2 * lh);
    acc0 = wmma4(a, b0, acc0);
    acc1 = wmma4(a, b1, acc1);
  }
  float v0[8], v1[8];
#pragma unroll
  for (int r = 0; r < 8; ++r) {
    int row = r + 8 * lh;
    const float* ar = addrow0 + (size_t)row * add_rs;
    v0[r] = acc0[r] + ar[n0a + lr];
    v1[r] = acc1[r] + ar[n0b + lr];
  }
  __syncthreads();  // all waves finished reading old sX
#pragma unroll
  for (int r = 0; r < 8; ++r) {
    int row = r + 8 * lh;
    sX[row * LDP + n0a + lr] = v0[r];
    sX[row * LDP + n0b + lr] = v1[r];
    if (outrow0) {
      float* orow = outrow0 + (size_t)row * out_rs;
      orow[n0a + lr] = v0[r];
      orow[n0b + lr] = v1[r];
    }
  }
  __syncthreads();  // new sX visible to all waves
}

// ---------------------------------------------------------------------------
// Pass 1: local scan per (chunk g, batch-tile bt) with zero carry; store only
// the chunk-end state E[g][b][:].
__global__ __launch_bounds__(256) void scan_pass1(float* __restrict__ E,
                                                  const float* __restrict__ U,
                                                  const float* __restrict__ Aeff) {
  __shared__ float sA[NS * LDP];  // 260 KB: A_eff, bank-conflict-free padded
  __shared__ float sX[BT * LDP];  // 16.6 KB: state
  const int g = blockIdx.x, bt = blockIdx.y;
  const int s = g * CHUNK;
  async_stage_matrix(Aeff, sA);
  for (int i = threadIdx.x; i < BT * NS; i += 256) {
    int r = i >> 8, n = i & 255;
    sX[r * LDP + n] = U[((size_t)(bt * BT + r) * TLEN + s) * NS + n];
  }
  __syncthreads();
  const size_t urs = (size_t)TLEN * NS;
  for (int i = 1; i < CHUNK; ++i) {
    const float* addb = U + ((size_t)(bt * BT) * TLEN + (s + i)) * NS;
    if (i + 1 < CHUNK) prefetch_add(addb + NS, urs);
    kf_step(sA, sX, addb, urs, nullptr, 0);
  }
  for (int i = threadIdx.x; i < BT * NS; i += 256) {
    int r = i >> 8, n = i & 255;
    E[((size_t)g * BATCH + bt * BT + r) * NS + n] = sX[r * LDP + n];
  }
}

// ---------------------------------------------------------------------------
// Carry chain: carry[0]=E[0]; carry[g] = carry[g-1] @ (A_eff^CHUNK)^T + E[g].
// Batch rows independent -> one workgroup per 16-row batch tile.
__global__ __launch_bounds__(256) void carry_chain(float* __restrict__ Carry,
                                                   const float* __restrict__ E,
                                                   const float* __restrict__ Pc) {
  __shared__ float sA[NS * LDP];
  __shared__ float sX[BT * LDP];
  const int bt = blockIdx.x;
  async_stage_matrix(Pc, sA);
  for (int i = threadIdx.x; i < BT * NS; i += 256) {
    int r = i >> 8, n = i & 255;
    float v = E[((size_t)(bt * BT + r)) * NS + n];  // g = 0
    sX[r * LDP + n] = v;
    Carry[((size_t)(bt * BT + r)) * NS + n] = v;
  }
  __syncthreads();
  for (int g = 1; g < NCH; ++g) {
    const float* addb = E + ((size_t)g * BATCH + bt * BT) * NS;
    float* outb = Carry + ((size_t)g * BATCH + bt * BT) * NS;
    if (g + 1 < NCH) prefetch_add(addb + (size_t)BATCH * NS, NS);
    kf_step(sA, sX, addb, NS, outb, NS);
  }
}

// ---------------------------------------------------------------------------
// Pass 2: rerun each chunk with its true carry-in, overwriting U (=d_out)
// in place with x_t.  Read-before-write within each step is safe: each (b,n)
// element is owned by exactly one lane.
__global__ __launch_bounds__(256) void scan_pass2(float* __restrict__ OutU,
                                                  const float* __restrict__ Carry,
                                                  const float* __restrict__ Aeff) {
  __shared__ float sA[NS * LDP];
  __shared__ float sX[BT * LDP];
  const int g = blockIdx.x, bt = blockIdx.y;
  const int s = g * CHUNK;
  async_stage_matrix(Aeff, sA);
  for (int i = threadIdx.x; i < BT * NS; i += 256) {
    int r = i >> 8, n = i & 255;
    float v = 0.f;
    if (g > 0) v = Carry[((size_t)(g - 1) * BATCH + bt * BT + r) * NS + n];
    sX[r * LDP + n] = v;
  }
  __syncthreads();
  const size_t urs = (size_t)TLEN * NS;
  for (int i = 0; i < CHUNK; ++i) {
    float* base = OutU + ((size_t)(bt * BT) * TLEN + (s + i)) * NS;
    if (i + 1 < CHUNK) prefetch_add(base + NS, urs);
    kf_step(sA, sX, base, urs, base, urs);
  }
}

// ---------------------------------------------------------------------------
extern "C" void kernel_launch(void* const* d_in, const int* in_sizes, int n_in,
                              void* d_out, int out_size, void* d_ws, size_t ws_size,
                              hipStream_t stream) {
  const float* y  = (const float*)d_in[0];   // [B, T, NO]
  const float* A  = (const float*)d_in[1];   // [NS, NS]
  const float* H  = (const float*)d_in[2];   // [NO, NS]
  const float* Km = (const float*)d_in[3];   // [NS, NO]
  float* U = (float*)d_out;                  // [B*T, NS]; holds u then x

  float* w    = (float*)d_ws;                // ~5.9 MB total
  float* T1   = w;                           // [NO, NS] = H @ A
  float* Aeff = T1 + NO * NS;                // [NS, NS]
  float* P2   = Aeff + NS * NS;
  float* P4   = P2 + NS * NS;
  float* P8   = P4 + NS * NS;
  float* P16  = P8 + NS * NS;
  float* P32  = P16 + NS * NS;               // A_eff^CHUNK
  float* E    = P32 + NS * NS;               // [NCH, BATCH, NS]
  float* Cr   = E + (size_t)NCH * BATCH * NS;

  // A_eff = A - K @ (H @ A); then log-double to A_eff^32.
  gemm_naive<<<(NO * NS + 255) / 256, 256, 0, stream>>>(T1, H, A, NO, NS, NS, 1.f, nullptr);
  gemm_naive<<<(NS * NS + 255) / 256, 256, 0, stream>>>(Aeff, Km, T1, NS, NS, NO, -1.f, A);
  gemm_naive<<<(NS * NS + 255) / 256, 256, 0, stream>>>(P2, Aeff, Aeff, NS, NS, NS, 1.f, nullptr);
  gemm_naive<<<(NS * NS + 255) / 256, 256, 0, stream>>>(P4, P2, P2, NS, NS, NS, 1.f, nullptr);
  gemm_naive<<<(NS * NS + 255) / 256, 256, 0, stream>>>(P8, P4, P4, NS, NS, NS, 1.f, nullptr);
  gemm_naive<<<(NS * NS + 255) / 256, 256, 0, stream>>>(P16, P8, P8, NS, NS, NS, 1.f, nullptr);
  gemm_naive<<<(NS * NS + 255) / 256, 256, 0, stream>>>(P32, P16, P16, NS, NS, NS, 1.f, nullptr);

  // u_t = y_t @ K^T for all t, into d_out.
  proj_u<<<dim3(BATCH * TLEN / 16), 256, 0, stream>>>(U, y, Km);

  // Blocked parallel scan.
  scan_pass1<<<dim3(NCH, NBT), 256, 0, stream>>>(E, U, Aeff);
  carry_chain<<<dim3(NBT), 256, 0, stream>>>(Cr, E, P32);
  scan_pass2<<<dim3(NCH, NBT), 256, 0, stream>>>(U, Cr, Aeff);
}